// HGT_1906965479659
// MI455X (gfx1250) — compile-verified
//
#include <hip/hip_runtime.h>
#include <math.h>

#define HID 64
#define HEADS 4
#define DH 16

typedef __attribute__((ext_vector_type(2))) float v2f;
typedef __attribute__((ext_vector_type(8))) float v8f;

// ---------------------------------------------------------------- utilities

__global__ __launch_bounds__(256) void fill_u32_kernel(unsigned int* __restrict__ p,
                                                       unsigned int v, int n) {
  int i = blockIdx.x * blockDim.x + threadIdx.x;
  if (i < n) p[i] = v;
}

__device__ __forceinline__ int fordered(float f) {
  int i = __float_as_int(f);
  return (i >= 0) ? i : (i ^ 0x7fffffff);
}
__device__ __forceinline__ float funordered(int i) {
  return __int_as_float((i >= 0) ? i : (i ^ 0x7fffffff));
}

// ------------------------------------------------------------- WMMA GEMM
// C[n,64] = act(A[n,K] @ W[K,64] + bias), K in {16,32,64}, n % 16 == 0.
// One wave per 16-row tile; 4 fp32 WMMA accumulators cover all 64 columns;
// the A fragment is reused across the 4 column tiles each K-step.
__global__ __launch_bounds__(256) void gemm64_wmma(
    const float* __restrict__ A, const float* __restrict__ W,
    const float* __restrict__ bias, float* __restrict__ C,
    int nRows, int K, int act)
{
  int wave = blockIdx.x * (blockDim.x >> 5) + (threadIdx.x >> 5);
  int nTiles = nRows >> 4;
  if (wave >= nTiles) return;       // wave-uniform: surviving waves keep EXEC all-ones
  int lane = threadIdx.x & 31;
  int rlo  = lane & 15;             // column within 16-wide tile / row within A tile
  int up   = lane >> 4;             // 0 or 1 (which half of the K-quad / M-half)
  int khalf = up << 1;              // 0 or 2

  const float* Arow = A + (size_t)((wave << 4) + rlo) * (size_t)K;

  v8f c0 = {0.f,0.f,0.f,0.f,0.f,0.f,0.f,0.f};
  v8f c1 = c0, c2 = c0, c3 = c0;

  for (int kk = 0; kk < K; kk += 4) {
    int ka = kk + khalf;
    v2f a;
    a.x = Arow[ka];
    a.y = Arow[ka + 1];
    const float* w0 = W + (size_t)ka * HID + rlo;   // row ka of W, this lane's column
    v2f b0, b1, b2, b3;
    b0.x = w0[0];       b0.y = w0[HID];
    b1.x = w0[16];      b1.y = w0[HID + 16];
    b2.x = w0[32];      b2.y = w0[HID + 32];
    b3.x = w0[48];      b3.y = w0[HID + 48];
    c0 = __builtin_amdgcn_wmma_f32_16x16x4_f32(false, a, false, b0, (short)0, c0, false, false);
    c1 = __builtin_amdgcn_wmma_f32_16x16x4_f32(false, a, false, b1, (short)0, c1, false, false);
    c2 = __builtin_amdgcn_wmma_f32_16x16x4_f32(false, a, false, b2, (short)0, c2, false, false);
    c3 = __builtin_amdgcn_wmma_f32_16x16x4_f32(false, a, false, b3, (short)0, c3, false, false);
  }

  float bv0 = bias[rlo], bv1 = bias[16 + rlo], bv2 = bias[32 + rlo], bv3 = bias[48 + rlo];
  int rbase = (wave << 4) + (up << 3);
#pragma unroll
  for (int j = 0; j < 8; ++j) {
    float* crow = C + (size_t)(rbase + j) * HID + rlo;
    float x0 = c0[j] + bv0, x1 = c1[j] + bv1, x2 = c2[j] + bv2, x3 = c3[j] + bv3;
    if (act) {
      x0 = fmaxf(x0, 0.f); x1 = fmaxf(x1, 0.f);
      x2 = fmaxf(x2, 0.f); x3 = fmaxf(x3, 0.f);
    }
    crow[0] = x0; crow[16] = x1; crow[32] = x2; crow[48] = x3;
  }
}

// ----------------------------------------------------------- edge kernels
// Pass 1: alpha[e,h] = (q[dst,h,:] . (k[src,h,:] @ A[h])) * p[h] / sqrt(DH);
//         running segment max via ordered-int atomicMax.
__global__ __launch_bounds__(256) void edge_alpha_kernel(
    const int* __restrict__ src, const int* __restrict__ dst,
    const float* __restrict__ Kt, const float* __restrict__ Qt,
    const float* __restrict__ Arel, const float* __restrict__ Prel,
    float* __restrict__ alpha, int* __restrict__ mBuf, int E)
{
  int t = blockIdx.x * blockDim.x + threadIdx.x;
  if (t >= E * HEADS) return;
  int e = t >> 2, h = t & 3;
  int s = src[e], d = dst[e];
  const float* kp = Kt + (size_t)s * HID + h * DH;
  const float* qp = Qt + (size_t)d * HID + h * DH;
  const float* Ap = Arel + h * DH * DH;
  float kv[DH], qv[DH];
#pragma unroll
  for (int i = 0; i < DH; ++i) { kv[i] = kp[i]; qv[i] = qp[i]; }
  float acc = 0.f;
#pragma unroll
  for (int f = 0; f < DH; ++f) {
    float ke = 0.f;
#pragma unroll
    for (int dd = 0; dd < DH; ++dd) ke += kv[dd] * Ap[dd * DH + f];
    acc += qv[f] * ke;
  }
  float al = acc * Prel[h] * 0.25f;   // 1/sqrt(16)
  alpha[(size_t)e * HEADS + h] = al;
  atomicMax(mBuf + (size_t)d * HEADS + h, fordered(al));
}

// Pass 2: e = exp(alpha - m[dst]); alpha <- e; segment sum via atomicAdd.
__global__ __launch_bounds__(256) void edge_expsum_kernel(
    const int* __restrict__ dst, float* __restrict__ alpha,
    const int* __restrict__ mBuf, float* __restrict__ sBuf, int E)
{
  int t = blockIdx.x * blockDim.x + threadIdx.x;
  if (t >= E * HEADS) return;
  int e = t >> 2, h = t & 3;
  int d = dst[e];
  float m = funordered(mBuf[(size_t)d * HEADS + h]);
  size_t idx = (size_t)e * HEADS + h;
  float ex = expf(alpha[idx] - m);
  alpha[idx] = ex;
  atomicAdd(sBuf + (size_t)d * HEADS + h, ex);
}

// Pass 3: w = e/(s[dst]+1e-16); v_e = v[src,h,:] @ M[h]; agg[dst] += w*v_e.
__global__ __launch_bounds__(256) void edge_agg_kernel(
    const int* __restrict__ src, const int* __restrict__ dst,
    const float* __restrict__ Vt, const float* __restrict__ Mrel,
    const float* __restrict__ alpha, const float* __restrict__ sBuf,
    float* __restrict__ agg, int E)
{
  int t = blockIdx.x * blockDim.x + threadIdx.x;
  if (t >= E * HEADS) return;
  int e = t >> 2, h = t & 3;
  int s = src[e], d = dst[e];
  float w = alpha[(size_t)e * HEADS + h] / (sBuf[(size_t)d * HEADS + h] + 1e-16f);
  const float* vp = Vt + (size_t)s * HID + h * DH;
  const float* Mp = Mrel + h * DH * DH;
  float vv[DH];
#pragma unroll
  for (int i = 0; i < DH; ++i) vv[i] = vp[i];
  float* ap = agg + (size_t)d * HID + h * DH;
#pragma unroll
  for (int f = 0; f < DH; ++f) {
    float ve = 0.f;
#pragma unroll
    for (int dd = 0; dd < DH; ++dd) ve += vv[dd] * Mp[dd * DH + f];
    atomicAdd(ap + f, w * ve);
  }
}

// ---------------------------------------------------------- node epilogues

__global__ __launch_bounds__(256) void gelu_kernel(float* __restrict__ x, int n) {
  int i = blockIdx.x * blockDim.x + threadIdx.x;
  if (i >= n) return;
  float v = x[i];
  float t = tanhf(0.7978845608028654f * (v + 0.044715f * v * v * v));
  x[i] = 0.5f * v * (1.f + t);
}

__global__ __launch_bounds__(256) void gate_relu_kernel(
    float* __restrict__ h, const float* __restrict__ o,
    const float* __restrict__ skip, int n)
{
  int i = blockIdx.x * blockDim.x + threadIdx.x;
  if (i >= n) return;
  float g = 1.f / (1.f + expf(-skip[0]));
  float v = g * o[i] + (1.f - g) * h[i];
  h[i] = fmaxf(v, 0.f);
}

__global__ __launch_bounds__(256) void head_out_kernel(
    const float* __restrict__ H, const float* __restrict__ W2,
    const float* __restrict__ b2, float* __restrict__ out, int n)
{
  int i = blockIdx.x * blockDim.x + threadIdx.x;
  if (i >= n) return;
  const float* r = H + (size_t)i * HID;
  float acc = 0.f;
#pragma unroll
  for (int c = 0; c < HID; ++c) acc += r[c] * W2[c];
  out[i] = acc + b2[0];
}

// ----------------------------------------------------------------- driver

extern "C" void kernel_launch(void* const* d_in, const int* in_sizes, int n_in,
                              void* d_out, int out_size, void* d_ws, size_t ws_size,
                              hipStream_t stream)
{
  (void)in_sizes; (void)n_in; (void)out_size; (void)ws_size;
  const int NF = 250000, NA = 2000;
  const int Edep = 250000, Earr = 250000, Enext = 500000;

  const float* x_f      = (const float*)d_in[0];
  const float* x_a      = (const float*)d_in[1];
  const int*   dep_src  = (const int*)d_in[2];
  const int*   dep_dst  = (const int*)d_in[3];
  const int*   arr_src  = (const int*)d_in[4];
  const int*   arr_dst  = (const int*)d_in[5];
  const int*   next_src = (const int*)d_in[6];
  const int*   next_dst = (const int*)d_in[7];

  // params flattened as a JAX pytree (dict keys sorted alphabetically)
  const float* head_W1 = (const float*)d_in[8];
  const float* head_b1 = (const float*)d_in[9];
  const float* head_W2 = (const float*)d_in[10];
  const float* head_b2 = (const float*)d_in[11];
  const float* inW_a   = (const float*)d_in[12];
  const float* inW_f   = (const float*)d_in[13];
  const float* inb_a   = (const float*)d_in[14];
  const float* inb_f   = (const float*)d_in[15];

  struct LP {
    const float *a_arr, *a_dep, *a_next;
    const float *kW_a, *kW_f, *kb_a, *kb_f;
    const float *m_arr, *m_dep, *m_next;
    const float *oW_a, *oW_f, *ob_a, *ob_f;
    const float *p_arr, *p_dep, *p_next;
    const float *qW_a, *qW_f, *qb_a, *qb_f;
    const float *skip_a, *skip_f;
    const float *vW_a, *vW_f, *vb_a, *vb_f;
  } L[2];
  for (int l = 0; l < 2; ++l) {
    int b = 16 + 27 * l;
    const float** pp = (const float**)&L[l];
    for (int j = 0; j < 27; ++j) pp[j] = (const float*)d_in[b + j];
  }

  // workspace carve-up (floats); ~347 MB total
  float* ws = (float*)d_ws;
  size_t off = 0;
  auto alloc = [&](size_t n) { float* p = ws + off; off += n; return p; };
  float* h_f     = alloc((size_t)NF * HID);
  float* k_f     = alloc((size_t)NF * HID);   // also reused as o_f / attn scratch
  float* q_f     = alloc((size_t)NF * HID);   // also reused as head hidden
  float* v_f     = alloc((size_t)NF * HID);
  float* agg_f   = alloc((size_t)NF * HID);
  float* h_a     = alloc((size_t)NA * HID);
  float* k_a     = alloc((size_t)NA * HID);
  float* q_a     = alloc((size_t)NA * HID);
  float* v_a     = alloc((size_t)NA * HID);
  float* agg_a   = alloc((size_t)NA * HID);
  float* al_dep  = alloc((size_t)Edep  * HEADS);
  float* al_arr  = alloc((size_t)Earr  * HEADS);
  float* al_next = alloc((size_t)Enext * HEADS);
  int*   m_f     = (int*)alloc((size_t)NF * HEADS);
  float* s_f     = alloc((size_t)NF * HEADS);
  int*   m_a     = (int*)alloc((size_t)NA * HEADS);
  float* s_a     = alloc((size_t)NA * HEADS);

  auto GEMM = [&](const float* A, const float* W, const float* bias, float* C,
                  int n, int K, int act) {
    int tiles = n >> 4;               // n is a multiple of 16 for all calls
    int blocks = (tiles + 7) >> 3;    // 8 waves per 256-thread block
    gemm64_wmma<<<blocks, 256, 0, stream>>>(A, W, bias, C, n, K, act);
  };
  auto FILL = [&](void* p, unsigned v, size_t n) {
    int blocks = (int)((n + 255) / 256);
    fill_u32_kernel<<<blocks, 256, 0, stream>>>((unsigned int*)p, v, (int)n);
  };
  auto E1 = [&](const int* s, const int* d, const float* Kt, const float* Qt,
                const float* Ar, const float* Pr, float* al, int* mB, int E) {
    edge_alpha_kernel<<<(E * HEADS + 255) / 256, 256, 0, stream>>>(s, d, Kt, Qt, Ar, Pr, al, mB, E);
  };
  auto E2 = [&](const int* d, float* al, const int* mB, float* sB, int E) {
    edge_expsum_kernel<<<(E * HEADS + 255) / 256, 256, 0, stream>>>(d, al, mB, sB, E);
  };
  auto E3 = [&](const int* s, const int* d, const float* Vt, const float* Mr,
                const float* al, const float* sB, float* ag, int E) {
    edge_agg_kernel<<<(E * HEADS + 255) / 256, 256, 0, stream>>>(s, d, Vt, Mr, al, sB, ag, E);
  };

  // ---- input projection + relu
  GEMM(x_f, inW_f, inb_f, h_f, NF, 32, 1);
  GEMM(x_a, inW_a, inb_a, h_a, NA, 16, 1);

  // ---- 2 HGT layers
  for (int l = 0; l < 2; ++l) {
    LP& P = L[l];
    GEMM(h_f, P.kW_f, P.kb_f, k_f, NF, HID, 0);
    GEMM(h_f, P.qW_f, P.qb_f, q_f, NF, HID, 0);
    GEMM(h_f, P.vW_f, P.vb_f, v_f, NF, HID, 0);
    GEMM(h_a, P.kW_a, P.kb_a, k_a, NA, HID, 0);
    GEMM(h_a, P.qW_a, P.qb_a, q_a, NA, HID, 0);
    GEMM(h_a, P.vW_a, P.vb_a, v_a, NA, HID, 0);

    FILL(m_f, 0x80000000u, (size_t)NF * HEADS);  // ordered-int -inf
    FILL(s_f, 0u, (size_t)NF * HEADS);
    FILL(agg_f, 0u, (size_t)NF * HID);
    FILL(m_a, 0x80000000u, (size_t)NA * HEADS);
    FILL(s_a, 0u, (size_t)NA * HEADS);
    FILL(agg_a, 0u, (size_t)NA * HID);

    // dep: flight -> airport; arr: airport -> flight; next: flight -> flight
    E1(dep_src,  dep_dst,  k_f, q_a, P.a_dep,  P.p_dep,  al_dep,  m_a, Edep);
    E1(arr_src,  arr_dst,  k_a, q_f, P.a_arr,  P.p_arr,  al_arr,  m_f, Earr);
    E1(next_src, next_dst, k_f, q_f, P.a_next, P.p_next, al_next, m_f, Enext);

    E2(dep_dst,  al_dep,  m_a, s_a, Edep);
    E2(arr_dst,  al_arr,  m_f, s_f, Earr);
    E2(next_dst, al_next, m_f, s_f, Enext);

    E3(dep_src,  dep_dst,  v_f, P.m_dep,  al_dep,  s_a, agg_a, Edep);
    E3(arr_src,  arr_dst,  v_a, P.m_arr,  al_arr,  s_f, agg_f, Earr);
    E3(next_src, next_dst, v_f, P.m_next, al_next, s_f, agg_f, Enext);

    gelu_kernel<<<(NF * HID + 255) / 256, 256, 0, stream>>>(agg_f, NF * HID);
    gelu_kernel<<<(NA * HID + 255) / 256, 256, 0, stream>>>(agg_a, NA * HID);

    GEMM(agg_f, P.oW_f, P.ob_f, k_f, NF, HID, 0);   // o_f into k_f (free now)
    GEMM(agg_a, P.oW_a, P.ob_a, k_a, NA, HID, 0);

    gate_relu_kernel<<<(NF * HID + 255) / 256, 256, 0, stream>>>(h_f, k_f, P.skip_f, NF * HID);
    gate_relu_kernel<<<(NA * HID + 255) / 256, 256, 0, stream>>>(h_a, k_a, P.skip_a, NA * HID);
  }

  // ---- head
  GEMM(h_f, head_W1, head_b1, q_f, NF, HID, 1);
  head_out_kernel<<<(NF + 255) / 256, 256, 0, stream>>>(q_f, head_W2, head_b2,
                                                        (float*)d_out, NF);
}